// GINElayer_78151224918242
// MI455X (gfx1250) — compile-verified
//
#include <hip/hip_runtime.h>

typedef float v2f __attribute__((ext_vector_type(2)));
typedef float v4f __attribute__((ext_vector_type(4)));
typedef float v8f __attribute__((ext_vector_type(8)));

#define N_NODES 100000
#define N_EDGES 1600000
#define D_FEAT  48

// ---------------------------------------------------------------------------
// Kernel 1: out = (1 + eps) * node_feat   (also initializes the scatter target)
// 4.8M floats -> 1.2M float4 chunks, pure b128 streaming.
// ---------------------------------------------------------------------------
__global__ void gine_init_kernel(const float* __restrict__ node,
                                 const float* __restrict__ eps,
                                 float* __restrict__ out) {
    const int total = N_NODES * (D_FEAT / 4);
    int idx = blockIdx.x * blockDim.x + threadIdx.x;
    if (idx >= total) return;
    const float s = 1.0f + eps[0];
    v4f v = ((const v4f*)node)[idx];
    ((v4f*)out)[idx] = v * s;
}

// ---------------------------------------------------------------------------
// Kernel 2: one wave32 handles a tile of 16 edges.
//   M(16 edges x 48 feats) = relu(node[src] + edge)            (VALU)
//   S(16x16)               = one-hot dst-equality matrix        (shuffles)
//   OUT(16x16 per f-chunk) = S x M_chunk  via 4x V_WMMA_F32_16X16X4_F32
//   scatter: one global_atomic_add_f32 per (first-occurrence row, column)
// Dedup via ballot ensures duplicate dst rows (which WMMA fills with the full
// per-dst sum) are emitted exactly once.
// ---------------------------------------------------------------------------
__global__ void __launch_bounds__(256)
gine_edge_wmma_kernel(const float* __restrict__ node,
                      const float* __restrict__ edge,
                      const int*  __restrict__ src,
                      const int*  __restrict__ dst,
                      float* __restrict__ out) {
    const int lane = threadIdx.x & 31;
    const int m    = lane & 15;      // row index owned by this lane (0..15)
    const int half = lane >> 4;      // 0: lanes 0-15, 1: lanes 16-31
    const int wave = (blockIdx.x * blockDim.x + threadIdx.x) >> 5;
    const int nTiles = N_EDGES / 16;
    if (wave >= nTiles) return;      // wave-uniform; active waves keep EXEC=all-1s
    const int eBase = wave * 16;

    const int d_m = dst[eBase + m];
    const int s_m = src[eBase + m];

    // Run ahead on the 307MB edge stream (global_prefetch_b8, RT speculative)
    __builtin_prefetch(edge + (size_t)(eBase + 256) * D_FEAT + lane * 24, 0, 0);

    // ---- first-occurrence mask over the 16 dst values (dedup rows) ----
    unsigned matchmask = 0u;
#pragma unroll
    for (int j = 0; j < 16; ++j) {
        int dj = __shfl(d_m, j, 32);
        matchmask |= (dj == d_m) ? (1u << j) : 0u;
    }
    const bool first = ((__ffs(matchmask) - 1) == m);
    const unsigned long long bal = __ballot(first);

    // ---- A = one-hot scatter matrix, 16x4 f32 per WMMA (ISA 7.12.2 layout:
    //      VGPR0 holds K = half?2:0, VGPR1 holds K = half?3:1, M = lane&15) ----
    v2f A[4];
#pragma unroll
    for (int kc = 0; kc < 4; ++kc) {
        const int k0 = 4 * kc + (half ? 2 : 0);
        const int k1 = k0 + 1;
        const int dk0 = __shfl(d_m, k0, 32);
        const int dk1 = __shfl(d_m, k1, 32);
        A[kc][0] = (dk0 == d_m) ? 1.0f : 0.0f;
        A[kc][1] = (dk1 == d_m) ? 1.0f : 0.0f;
    }

    // ---- 3 feature chunks of 16 columns each (48 = 3*16) ----
#pragma unroll
    for (int f = 0; f < 3; ++f) {
        v8f c = {};  // 16x16 f32 accumulator (8 VGPRs)
#pragma unroll
        for (int kc = 0; kc < 4; ++kc) {
            const int k0 = 4 * kc + (half ? 2 : 0);
            const int k1 = k0 + 1;
            const int e0 = eBase + k0;
            const int e1 = eBase + k1;
            const int feat = f * 16 + m;          // column = lane&15
            const int s0 = __shfl(s_m, k0, 32);
            const int s1 = __shfl(s_m, k1, 32);
            // edge stream: read-once -> non-temporal (keep L2 for node/out)
            const float ev0 = __builtin_nontemporal_load(edge + (size_t)e0 * D_FEAT + feat);
            const float ev1 = __builtin_nontemporal_load(edge + (size_t)e1 * D_FEAT + feat);
            // node gather: 19.2MB table, L2-resident -> regular temporal
            const float nv0 = node[(size_t)s0 * D_FEAT + feat];
            const float nv1 = node[(size_t)s1 * D_FEAT + feat];
            v2f b;
            b[0] = fmaxf(nv0 + ev0, 0.0f);
            b[1] = fmaxf(nv1 + ev1, 0.0f);
            c = __builtin_amdgcn_wmma_f32_16x16x4_f32(
                    /*neg_a=*/false, A[kc], /*neg_b=*/false, b,
                    /*c_mod=*/(short)0, c, /*reuse_a=*/false, /*reuse_b=*/false);
        }
        // ---- scatter: C/D layout VGPR r -> row (r + 8*half), col = lane&15 ----
#pragma unroll
        for (int r = 0; r < 8; ++r) {
            const int i = r + 8 * half;
            const int di = __shfl(d_m, i, 32);   // shuffle with full EXEC, pre-branch
            if ((bal >> i) & 1ull) {
                atomicAdd(out + (size_t)di * D_FEAT + f * 16 + m, c[r]);
            }
        }
    }
}

// ---------------------------------------------------------------------------
extern "C" void kernel_launch(void* const* d_in, const int* in_sizes, int n_in,
                              void* d_out, int out_size, void* d_ws, size_t ws_size,
                              hipStream_t stream) {
    const float* node = (const float*)d_in[0];
    const float* edge = (const float*)d_in[1];
    const int*   src  = (const int*)d_in[2];
    const int*   dst  = (const int*)d_in[3];
    const float* eps  = (const float*)d_in[4];
    float* out = (float*)d_out;

    // Phase 1: out = (1+eps) * node_feat
    const int initTotal = N_NODES * (D_FEAT / 4);
    gine_init_kernel<<<(initTotal + 255) / 256, 256, 0, stream>>>(node, eps, out);

    // Phase 2: WMMA-based gather + scatter-add (16 edges per wave, 8 waves/block)
    const int nTiles = N_EDGES / 16;          // 100000 tiles == 100000 waves
    const int wavesPerBlock = 256 / 32;       // 8
    const int blocks = (nTiles + wavesPerBlock - 1) / wavesPerBlock;  // 12500
    gine_edge_wmma_kernel<<<blocks, 256, 0, stream>>>(node, edge, src, dst, out);
}